// AttentionLayer3D_79791902425126
// MI455X (gfx1250) — compile-verified
//
#include <hip/hip_runtime.h>

typedef __attribute__((ext_vector_type(2))) float v2f;
typedef __attribute__((ext_vector_type(8))) float v8f;

#define DD   24
#define HH   48
#define WWD  64
#define S    (DD * HH * WWD)     /* 73728  */
#define DP   26
#define HP   50
#define WP   66
#define SP   (DP * HP * WP)      /* 85800  */
#define CIN  32
#define COUT 48

// Load A-matrix (16x4 f32 slice of the 48x32 weight, rows mbase..mbase+15) for
// all 8 K-steps. Lane L<16 holds K = 4s+{0,1}, lane L>=16 holds K = 4s+{2,3}.
__device__ __forceinline__ void loadA(const float* __restrict__ Wm, int mbase,
                                      int lane, v2f a[8]) {
  const float* row = Wm + (mbase + (lane & 15)) * CIN + ((lane >> 4) << 1);
#pragma unroll
  for (int s = 0; s < 8; ++s) {
    a[s].x = row[4 * s];
    a[s].y = row[4 * s + 1];
  }
}

// ---------------- Phase 1a: q = Wq @ x  (M=48, N=73728, K=32) ----------------
__global__ void __launch_bounds__(128)
qkv_q_kernel(const float* __restrict__ x, const float* __restrict__ Wq,
             float* __restrict__ q) {
  const int lane  = threadIdx.x & 31;
  const int ntile = blockIdx.x * (blockDim.x >> 5) + (threadIdx.x >> 5);
  if (ntile >= S / 16) return;                  // uniform per wave
  const int col  = ntile * 16 + (lane & 15);
  const int koff = (lane >> 4) << 1;            // 0 or 2

  // B-matrix (4x16 f32) for all 8 K-steps: contiguous 16-wide column loads.
  v2f b[8];
  const float* xc = x + col;
#pragma unroll
  for (int s = 0; s < 8; ++s) {
    b[s].x = xc[(4 * s + koff) * S];
    b[s].y = xc[(4 * s + koff + 1) * S];
  }

#pragma unroll
  for (int mt = 0; mt < 3; ++mt) {
    v2f a[8];
    loadA(Wq, mt * 16, lane, a);
    v8f c = {};
#pragma unroll
    for (int s = 0; s < 8; ++s)
      c = __builtin_amdgcn_wmma_f32_16x16x4_f32(false, a[s], false, b[s],
                                                (short)0, c, false, false);
    const int mrow = mt * 16 + ((lane >> 4) << 3);
#pragma unroll
    for (int j = 0; j < 8; ++j) q[(mrow + j) * S + col] = c[j];
  }
}

// ------- Phase 1b: kf = Wk @ pad(x), vf = Wv @ pad(x) (N=85800, padded) ------
__global__ void __launch_bounds__(128)
qkv_kv_kernel(const float* __restrict__ x, const float* __restrict__ Wk,
              const float* __restrict__ Wv, float* __restrict__ kf,
              float* __restrict__ vf) {
  const int lane  = threadIdx.x & 31;
  const int ntile = blockIdx.x * (blockDim.x >> 5) + (threadIdx.x >> 5);
  const int NT    = (SP + 15) / 16;             // 5363 (last tile partial)
  if (ntile >= NT) return;                      // uniform per wave
  const int col  = ntile * 16 + (lane & 15);
  const int koff = (lane >> 4) << 1;

  // Fold zero-padding into the B load: map padded col -> interior x offset.
  int  srcoff = 0;
  bool ok     = false;
  if (col < SP) {
    const int dp = col / (HP * WP);
    const int r  = col - dp * (HP * WP);
    const int hp = r / WP;
    const int wp = r - hp * WP;
    if (dp >= 1 && dp <= DD && hp >= 1 && hp <= HH && wp >= 1 && wp <= WWD) {
      ok     = true;
      srcoff = (dp - 1) * (HH * WWD) + (hp - 1) * WWD + (wp - 1);
    }
  }
  const float* xs = x + srcoff;
  v2f b[8];
#pragma unroll
  for (int s = 0; s < 8; ++s) {
    b[s].x = ok ? xs[(4 * s + koff) * S] : 0.0f;
    b[s].y = ok ? xs[(4 * s + koff + 1) * S] : 0.0f;
  }

  const bool cstore = (col < SP);
#pragma unroll
  for (int mt = 0; mt < 3; ++mt) {
    const int mrow = mt * 16 + ((lane >> 4) << 3);
    {
      v2f a[8];
      loadA(Wk, mt * 16, lane, a);
      v8f c = {};
#pragma unroll
      for (int s = 0; s < 8; ++s)
        c = __builtin_amdgcn_wmma_f32_16x16x4_f32(false, a[s], false, b[s],
                                                  (short)0, c, false, false);
      if (cstore) {
#pragma unroll
        for (int j = 0; j < 8; ++j) kf[(mrow + j) * SP + col] = c[j];
      }
    }
    {
      v2f a[8];
      loadA(Wv, mt * 16, lane, a);
      v8f c = {};
#pragma unroll
      for (int s = 0; s < 8; ++s)
        c = __builtin_amdgcn_wmma_f32_16x16x4_f32(false, a[s], false, b[s],
                                                  (short)0, c, false, false);
      if (cstore) {
#pragma unroll
        for (int j = 0; j < 8; ++j) vf[(mrow + j) * SP + col] = c[j];
      }
    }
  }
}

// --------- Phase 2: per-(channel, voxel) 27-way softmax attention ------------
__global__ void __launch_bounds__(256)
attn_kernel(const float* __restrict__ q, const float* __restrict__ kf,
            const float* __restrict__ vf, const float* __restrict__ rel_h,
            const float* __restrict__ rel_w, const float* __restrict__ rel_d,
            float* __restrict__ out) {
  const int idx = blockIdx.x * blockDim.x + threadIdx.x;
  if (idx >= COUT * S) return;
  const int o = idx / S;
  const int p = idx - o * S;
  const int d = p / (HH * WWD);
  const int r = p - d * (HH * WWD);
  const int h = r / WWD;
  const int w = r - h * WWD;

  const float qv = q[idx];
  const int   g  = o >> 4;        // channel group: 0->rel_d(kw) 1->rel_h(kd) 2->rel_w(kh)
  const int   c3 = o & 15;
  const float* rel = (g == 0) ? rel_d : (g == 1) ? rel_h : rel_w;
  float bias3[3] = {rel[c3 * 3 + 0], rel[c3 * 3 + 1], rel[c3 * 3 + 2]};

  const float* kfo = kf + (size_t)o * SP;
  const float* vfo = vf + (size_t)o * SP;

  float lg[27];
  float mx = -3.4e38f;
#pragma unroll
  for (int kd = 0; kd < 3; ++kd)
#pragma unroll
    for (int kh = 0; kh < 3; ++kh)
#pragma unroll
      for (int kw = 0; kw < 3; ++kw) {
        const int j  = kd * 9 + kh * 3 + kw;
        const int pp = (d + kd) * (HP * WP) + (h + kh) * WP + (w + kw);
        const int bi = (g == 0) ? kw : (g == 1) ? kd : kh;
        const float l = qv * (kfo[pp] + bias3[bi]);
        lg[j] = l;
        mx    = fmaxf(mx, l);
      }

  float ssum = 0.0f;
#pragma unroll
  for (int j = 0; j < 27; ++j) {
    lg[j] = __expf(lg[j] - mx);
    ssum += lg[j];
  }

  float acc = 0.0f;
#pragma unroll
  for (int kd = 0; kd < 3; ++kd)
#pragma unroll
    for (int kh = 0; kh < 3; ++kh)
#pragma unroll
      for (int kw = 0; kw < 3; ++kw) {
        const int j  = kd * 9 + kh * 3 + kw;
        const int pp = (d + kd) * (HP * WP) + (h + kh) * WP + (w + kw);
        acc += lg[j] * vfo[pp];
      }
  out[idx] = acc / ssum;
}

extern "C" void kernel_launch(void* const* d_in, const int* in_sizes, int n_in,
                              void* d_out, int out_size, void* d_ws,
                              size_t ws_size, hipStream_t stream) {
  (void)in_sizes; (void)n_in; (void)out_size; (void)ws_size;
  const float* x     = (const float*)d_in[0];
  const float* Wq    = (const float*)d_in[1];
  const float* Wk    = (const float*)d_in[2];
  const float* Wv    = (const float*)d_in[3];
  const float* rel_h = (const float*)d_in[4];
  const float* rel_w = (const float*)d_in[5];
  const float* rel_d = (const float*)d_in[6];
  float* out = (float*)d_out;

  // Workspace layout: q (48*73728) | kf (48*85800) | vf (48*85800)  ~47 MB
  float* q  = (float*)d_ws;
  float* kf = q + (size_t)COUT * S;
  float* vf = kf + (size_t)COUT * SP;

  {  // q GEMM: 4608 N-tiles, 4 waves/block
    const int ntiles = S / 16;
    const int blocks = (ntiles + 3) / 4;
    qkv_q_kernel<<<blocks, 128, 0, stream>>>(x, Wq, q);
  }
  {  // k/v GEMM over padded volume: 5363 N-tiles
    const int ntiles = (SP + 15) / 16;
    const int blocks = (ntiles + 3) / 4;
    qkv_kv_kernel<<<blocks, 128, 0, stream>>>(x, Wk, Wv, kf, vf);
  }
  {  // fused window softmax-attention
    const int total = COUT * S;
    attn_kernel<<<(total + 255) / 256, 256, 0, stream>>>(q, kf, vf, rel_h,
                                                         rel_w, rel_d, out);
  }
}